// MarginCosineProduct_45483703665256
// MI455X (gfx1250) — compile-verified
//
#include <hip/hip_runtime.h>
#include <hip/hip_bf16.h>

typedef __attribute__((ext_vector_type(2))) float v2f;
typedef __attribute__((ext_vector_type(8))) float v8f;

#define BATCH   64
#define SEQ     512
#define DMODEL  768
#define NCAP    5
#define DCAP    10
#define NCOL    50            // NCAP*DCAP
#define KSTEPS  192           // DMODEL / 4
#define MROWS   (BATCH*SEQ)   // 32768

// workspace layout in floats
#define WS_U     0
#define WS_WPAD  (MROWS*NCOL)                 // 1,638,400
#define WS_FEAT  (WS_WPAD + 4*KSTEPS*64)      // +49,152
#define WS_W1    (WS_FEAT + BATCH*NCOL)       // +3,200

// ---------------------------------------------------------------------------
// Pack W [768,50] into per-lane WMMA B-operand layout, N padded to 64.
// Element (ntile, kstep, lane, j):
//   k = kstep*4 + ((lane>=16)?2:0) + j,  n = ntile*16 + (lane&15)
// stored contiguously so the GEMM loads one float2 (b64) per lane.
// ---------------------------------------------------------------------------
__global__ void build_wpad_kernel(const float* __restrict__ W,
                                  float* __restrict__ Wpad) {
    int idx = blockIdx.x * blockDim.x + threadIdx.x;   // 4*192*32 = 24576
    if (idx >= 4 * KSTEPS * 32) return;
    int lane = idx & 31;
    int ks   = (idx >> 5) % KSTEPS;
    int nt   = (idx >> 5) / KSTEPS;
    int n    = nt * 16 + (lane & 15);
    int kHalf = (lane >> 4) << 1;
#pragma unroll
    for (int j = 0; j < 2; ++j) {
        int k = ks * 4 + kHalf + j;
        float v = (n < NCOL) ? W[k * NCOL + n] : 0.0f;
        Wpad[((size_t)(nt * KSTEPS + ks) * 32 + lane) * 2 + j] = v;
    }
}

// ---------------------------------------------------------------------------
// u = hidden @ W  via V_WMMA_F32_16X16X4_F32.
// Each wave: 2 M-tiles (32 rows) x 4 N-tiles (N padded 50->64).
// A layout (16x4 f32): lanes 0-15 -> M=lane, K={0,1}; lanes 16-31 -> K={2,3}.
// ---------------------------------------------------------------------------
__global__ void __launch_bounds__(128)
capsule_gemm_kernel(const float* __restrict__ A,
                    const float* __restrict__ Bp,
                    float* __restrict__ U) {
    const int lane  = threadIdx.x & 31;
    const int wave  = blockIdx.x * (blockDim.x >> 5) + (threadIdx.x >> 5);
    const int mBase = wave * 32;
    const int mrow  = lane & 15;
    const int kHalf = (lane >> 4) << 1;              // 0 or 2

    const float* a0p = A + (size_t)(mBase + mrow) * DMODEL + kHalf;
    const float* a1p = a0p + (size_t)16 * DMODEL;
    const float* bp  = Bp + lane * 2;

    v8f acc[2][4];
#pragma unroll
    for (int t = 0; t < 2; ++t)
#pragma unroll
        for (int nt = 0; nt < 4; ++nt) acc[t][nt] = (v8f){0,0,0,0,0,0,0,0};

    for (int ks = 0; ks < KSTEPS; ++ks) {
        v2f a0 = *(const v2f*)(a0p + ks * 4);
        v2f a1 = *(const v2f*)(a1p + ks * 4);
#pragma unroll
        for (int nt = 0; nt < 4; ++nt) {
            v2f b = *(const v2f*)(bp + (size_t)(nt * KSTEPS + ks) * 64);
            acc[0][nt] = __builtin_amdgcn_wmma_f32_16x16x4_f32(
                false, a0, false, b, (short)0, acc[0][nt], false, false);
            acc[1][nt] = __builtin_amdgcn_wmma_f32_16x16x4_f32(
                false, a1, false, b, (short)0, acc[1][nt], false, false);
        }
    }

    // C/D layout: VGPR r: lanes 0-15 -> M=r, lanes 16-31 -> M=r+8; N=lane&15.
    const int colLo = lane & 15;
    const int rOff  = (lane >> 4) << 3;
#pragma unroll
    for (int t = 0; t < 2; ++t)
#pragma unroll
        for (int nt = 0; nt < 4; ++nt) {
            int col = nt * 16 + colLo;
            if (col < NCOL) {
#pragma unroll
                for (int r = 0; r < 8; ++r) {
                    int row = mBase + t * 16 + r + rOff;
                    U[(size_t)row * NCOL + col] = acc[t][nt][r];
                }
            }
        }
}

// ---------------------------------------------------------------------------
// Dynamic routing, 3 iterations. One 512-thread block per batch element;
// thread s holds u[b, s, :] (50 floats) in registers.
// ---------------------------------------------------------------------------
__device__ __forceinline__ float wave_sum(float v) {
#pragma unroll
    for (int off = 16; off > 0; off >>= 1) v += __shfl_xor(v, off, 32);
    return v;
}

__global__ void __launch_bounds__(512)
routing_kernel(const float* __restrict__ U,
               float* __restrict__ feat,
               float* __restrict__ w1) {
    __shared__ float accS[NCOL];
    __shared__ float outS[NCOL];
    const int b = blockIdx.x;
    const int s = threadIdx.x;

    float uv[NCOL];
    const float* up = U + ((size_t)b * SEQ + s) * NCOL;
#pragma unroll
    for (int i = 0; i < NCOL; ++i) uv[i] = up[i];

    float blog[NCAP] = {0.f, 0.f, 0.f, 0.f, 0.f};

    for (int it = 0; it < 3; ++it) {
        // softmax over the capsule axis (per (b,s))
        float m = blog[0];
#pragma unroll
        for (int n = 1; n < NCAP; ++n) m = fmaxf(m, blog[n]);
        float c[NCAP];
        float sum = 0.f;
#pragma unroll
        for (int n = 0; n < NCAP; ++n) { c[n] = expf(blog[n] - m); sum += c[n]; }
        float inv = 1.0f / sum;
#pragma unroll
        for (int n = 0; n < NCAP; ++n) c[n] *= inv;

        if (threadIdx.x < NCOL) accS[threadIdx.x] = 0.0f;
        __syncthreads();

        // outputs[n,d] = sum_s c[n,s] * u[s,n,d]  (wave reduce + LDS atomics)
#pragma unroll
        for (int i = 0; i < NCOL; ++i) {
            float v = c[i / DCAP] * uv[i];
            v = wave_sum(v);
            if ((threadIdx.x & 31) == 0) atomicAdd(&accS[i], v);
        }
        __syncthreads();

        // squash: x / sqrt(||x||^2 + 1e-7) over d
        if (threadIdx.x < NCAP) {
            int n = threadIdx.x;
            float s2 = 0.f;
#pragma unroll
            for (int d = 0; d < DCAP; ++d) {
                float v = accS[n * DCAP + d];
                s2 += v * v;
            }
            float invn = 1.0f / sqrtf(s2 + 1e-7f);
#pragma unroll
            for (int d = 0; d < DCAP; ++d)
                outS[n * DCAP + d] = accS[n * DCAP + d] * invn;
        }
        __syncthreads();

        if (it < 2) {
            // b[n,s] = sum_d outputs[n,d] * u[s,n,d]   (assignment per reference)
#pragma unroll
            for (int n = 0; n < NCAP; ++n) {
                float dot = 0.f;
#pragma unroll
                for (int d = 0; d < DCAP; ++d)
                    dot += outS[n * DCAP + d] * uv[n * DCAP + d];
                blog[n] = dot;
            }
        }
    }

    if (threadIdx.x < NCOL) feat[b * NCOL + threadIdx.x] = outS[threadIdx.x];
    if (threadIdx.x == 0) {
        float s2 = 0.f;
#pragma unroll
        for (int i = 0; i < NCOL; ++i) s2 += outS[i] * outS[i];
        w1[b] = sqrtf(s2);
    }
}

// ---------------------------------------------------------------------------
// Cosine head: ip = feat @ weight^T; cosine = ip / max(w1*w2, eps);
// output = 30*(cosine - onehot*0.15). d_out = [output | cosine | ip].
// ---------------------------------------------------------------------------
__global__ void final_kernel(const float* __restrict__ feat,
                             const float* __restrict__ w1,
                             const float* __restrict__ weight,
                             const int* __restrict__ label,
                             float* __restrict__ out) {
    const int b = blockIdx.x;
    const int c = threadIdx.x;
    if (c >= 150) return;

    float ip = 0.f, w2 = 0.f;
#pragma unroll
    for (int i = 0; i < NCOL; ++i) {
        float wv = weight[c * NCOL + i];
        ip += feat[b * NCOL + i] * wv;
        w2 += wv * wv;
    }
    float denom = fmaxf(w1[b] * sqrtf(w2), 1e-8f);
    float cosv  = ip / denom;
    float outv  = 30.0f * (cosv - ((c == label[b]) ? 0.15f : 0.0f));

    out[            b * 150 + c] = outv;
    out[ 64 * 150 + b * 150 + c] = cosv;
    out[128 * 150 + b * 150 + c] = ip;
}

// ---------------------------------------------------------------------------
extern "C" void kernel_launch(void* const* d_in, const int* in_sizes, int n_in,
                              void* d_out, int out_size, void* d_ws, size_t ws_size,
                              hipStream_t stream) {
    const float* hidden = (const float*)d_in[0];   // [64,512,768] f32
    const int*   label  = (const int*)  d_in[1];   // [64] i32
    const float* W      = (const float*)d_in[2];   // [1,768,50] f32
    const float* weight = (const float*)d_in[3];   // [150,50] f32
    float* out = (float*)d_out;                    // 3 * 64*150 f32
    float* ws  = (float*)d_ws;

    float* U    = ws + WS_U;
    float* Wpad = ws + WS_WPAD;
    float* feat = ws + WS_FEAT;
    float* w1   = ws + WS_W1;

    build_wpad_kernel<<<96, 256, 0, stream>>>(W, Wpad);
    capsule_gemm_kernel<<<256, 128, 0, stream>>>(hidden, Wpad, U);  // 1024 waves
    routing_kernel<<<BATCH, 512, 0, stream>>>(U, feat, w1);
    final_kernel<<<BATCH, 160, 0, stream>>>(feat, w1, weight, label, out);
}